// GNNCritic_4045859193497
// MI455X (gfx1250) — compile-verified
//
#include <hip/hip_runtime.h>
#include <hip/hip_bf16.h>
#include <math.h>

// ---------------------------------------------------------------------------
// GNN critic, fully fused: one workgroup per batch (A=128 agents).
//  - distance matrix + kNN (k=16) in LDS, 4 lanes/agent, shuffle-combined
//  - dense normalized adjacency M (128x128) built in LDS
//  - layer1: H1 = tanh(M @ (X@W1) + b1)       [WMMA f32 16x16x4, 16x64 strips]
//  - layer2: H2 = tanh(M @ (H1@W2) + b2)      [WMMA f32 16x16x4, 16x64 strips]
//  - out    = H2 @ Wout + bout                [4-lane shuffle reduction]
// HBM traffic: 8 MB obs in + 256 KB out  (~0.35 us floor @ 23.3 TB/s)
// ---------------------------------------------------------------------------

typedef float v2f __attribute__((ext_vector_type(2)));
typedef float v8f __attribute__((ext_vector_type(8)));

#define A_AGENTS 128
#define D_IN     32
#define HID      128
#define KNN      16
#define OBS_LD   36     // padded leading dim for obs tile (bank rotation)
#define MAT_LD   132    // padded leading dim for 128-wide matrices
#define NTHREADS 512    // 16 waves (wave32)
#define NWAVES   (NTHREADS / 32)
#define BIGF     3.4e38f

// LDS layout (in floats)
#define OFF_OBS  0
#define OFF_M    (OFF_OBS + A_AGENTS * OBS_LD)
#define OFF_T    (OFF_M   + A_AGENTS * MAT_LD)   // distances, then X@W / H@W
#define OFF_H    (OFF_T   + A_AGENTS * MAT_LD)
#define OFF_DINV (OFF_H   + A_AGENTS * MAT_LD)
#define LDS_FLOATS (OFF_DINV + A_AGENTS)
#define LDS_BYTES  (LDS_FLOATS * 4)

// Hardware tanh (V_TANH_F32) when the builtin exists; ocml fallback otherwise.
__device__ __forceinline__ float fast_tanh(float x) {
#if __has_builtin(__builtin_amdgcn_tanhf)
    return __builtin_amdgcn_tanhf(x);
#elif __has_builtin(__builtin_amdgcn_tanh_f32)
    return __builtin_amdgcn_tanh_f32(x);
#else
    return tanhf(x);
#endif
}

// One 16x64 strip (4 tiles, shared A fragment) of OUT = A(.,K) @ B(K,.),
// fp32 WMMA 16x16x4, K multiple of 4. A row-major (lda), B row-major (ldb).
// Optional bias+tanh epilogue.
template<bool ACT>
__device__ __forceinline__ void strip_mm16x4(const float* __restrict__ A, int lda,
                                             const float* __restrict__ B, int ldb,
                                             float* __restrict__ OUT, int ldo,
                                             const float* __restrict__ bias,
                                             int K, int tm, int tn0, int lane)
{
    const int g = lane >> 4;    // 0: k in {k0,k0+1}; 1: k in {k0+2,k0+3}
    const int r = lane & 15;    // A: row within tile; B: col within tile
    const float* Arow = A + (tm * 16 + r) * lda;
    const float* Bcol = B + tn0 * 16 + r;
    v8f c0 = {}, c1 = {}, c2 = {}, c3 = {};
    for (int k0 = 0; k0 < K; k0 += 4) {
        const int ka = k0 + 2 * g;
        const float* Brow0 = Bcol + (ka)     * ldb;
        const float* Brow1 = Bcol + (ka + 1) * ldb;
        v2f a, b0, b1, b2, b3;
        a.x  = Arow[ka];   a.y  = Arow[ka + 1];
        b0.x = Brow0[0];   b0.y = Brow1[0];
        b1.x = Brow0[16];  b1.y = Brow1[16];
        b2.x = Brow0[32];  b2.y = Brow1[32];
        b3.x = Brow0[48];  b3.y = Brow1[48];
        c0 = __builtin_amdgcn_wmma_f32_16x16x4_f32(false, a, false, b0, (short)0, c0, false, false);
        c1 = __builtin_amdgcn_wmma_f32_16x16x4_f32(false, a, false, b1, (short)0, c1, false, false);
        c2 = __builtin_amdgcn_wmma_f32_16x16x4_f32(false, a, false, b2, (short)0, c2, false, false);
        c3 = __builtin_amdgcn_wmma_f32_16x16x4_f32(false, a, false, b3, (short)0, c3, false, false);
    }
    // C/D layout: row = tm*16 + 8*g + v, col = tn0*16 + t*16 + r
    const v8f cc[4] = {c0, c1, c2, c3};
    float* obase = OUT + (tm * 16 + 8 * g) * ldo + tn0 * 16 + r;
#pragma unroll
    for (int t = 0; t < 4; ++t) {
        const float bval = ACT ? bias[tn0 * 16 + t * 16 + r] : 0.0f;
#pragma unroll
        for (int v = 0; v < 8; ++v) {
            float val = cc[t][v];
            if (ACT) val = fast_tanh(val + bval);
            obase[v * ldo + t * 16] = val;
        }
    }
}

__global__ void __launch_bounds__(NTHREADS)
gnn_critic_fused(const float* __restrict__ obs,
                 const float* __restrict__ W1, const float* __restrict__ b1,
                 const float* __restrict__ W2, const float* __restrict__ b2,
                 const float* __restrict__ Wout, const float* __restrict__ bout,
                 float* __restrict__ out)
{
    extern __shared__ float smem[];
    float* obs_s  = smem + OFF_OBS;   // [128][36], cols 0..31 valid
    float* M_s    = smem + OFF_M;     // [128][132] normalized adjacency
    float* T_s    = smem + OFF_T;     // [128][132] dist matrix, then X@W
    float* H_s    = smem + OFF_H;     // [128][132] hidden activations
    float* dinv_s = smem + OFF_DINV;  // [128]

    const int tid  = threadIdx.x;
    const int lane = tid & 31;
    const int wave = tid >> 5;         // 0..15
    const int tm   = wave >> 1;        // strip row   0..7
    const int tn0  = (wave & 1) * 4;   // strip col0  0 or 4
    const int bidx = blockIdx.x;

    // ---- Stage 0: load obs batch (128x32) into padded LDS; zero M ----------
    const float* gobs = obs + (size_t)bidx * (A_AGENTS * D_IN);
    for (int i = tid; i < A_AGENTS * (D_IN / 4); i += NTHREADS) {
        const int row = i / (D_IN / 4);
        const int c4  = i % (D_IN / 4);
        const float4 v = ((const float4*)gobs)[i];
        float* dst = obs_s + row * OBS_LD + c4 * 4;
        dst[0] = v.x; dst[1] = v.y; dst[2] = v.z; dst[3] = v.w;
    }
    for (int i = tid; i < A_AGENTS * MAT_LD; i += NTHREADS) M_s[i] = 0.0f;
    __syncthreads();

    // ---- Stage 1: pairwise squared distances into T_s (self = +inf) -------
    for (int e = tid; e < A_AGENTS * A_AGENTS; e += NTHREADS) {
        const int i = e >> 7;
        const int j = e & 127;
        const float dx = obs_s[i * OBS_LD + 0] - obs_s[j * OBS_LD + 0];
        const float dy = obs_s[i * OBS_LD + 1] - obs_s[j * OBS_LD + 1];
        float d = dx * dx + dy * dy;
        if (i == j) d = BIGF;                       // exclude self
        T_s[i * MAT_LD + j] = d;
    }
    __syncthreads();

    // ---- Stage 2: kNN selection, 4 lanes per agent, no intra-round barriers
    {
        const int a  = tid >> 2;                    // agent 0..127
        const int sl = tid & 3;                     // slice 0..3 (same wave)
        float* drow = T_s + a * MAT_LD + sl * 32;
        for (int t = 0; t < KNN; ++t) {
            float best = BIGF;
            int bestj = 0;
            for (int j = 0; j < 32; ++j) {          // ascending: ties -> low j
                const float d = drow[j];
                if (d < best) { best = d; bestj = sl * 32 + j; }
            }
            // lexicographic min over the 4 slices (dist, then lower index)
            for (int off = 1; off < 4; off <<= 1) {
                const float ob = __shfl_xor(best, off);
                const int   oj = __shfl_xor(bestj, off);
                if (ob < best || (ob == best && oj < bestj)) {
                    best = ob; bestj = oj;
                }
            }
            if (sl == 0) {
                T_s[a * MAT_LD + bestj] = BIGF;     // exclude for next round
                M_s[bestj * MAT_LD + a] = 1.0f;     // edge a -> bestj
            }
            // same-wave LDS ops are in-order: next round sees the writes
        }
    }
    __syncthreads();

    // ---- Stage 3: degree (1 + incoming) -> dinv = rsqrt(deg) --------------
    {
        const int a  = tid >> 2;
        const int sl = tid & 3;
        const float* seg = M_s + a * MAT_LD + sl * 32;
        float s = 0.0f;
        for (int i = 0; i < 32; ++i) s += seg[i];
        s += __shfl_xor(s, 1);
        s += __shfl_xor(s, 2);
        if (sl == 0) dinv_s[a] = rsqrtf(1.0f + s);
    }
    __syncthreads();

    // ---- Stage 4: M[j][i] = adj * d_i*d_j ; diag = d_j^2 ------------------
    for (int e = tid; e < A_AGENTS * A_AGENTS; e += NTHREADS) {
        const int j = e >> 7;
        const int i = e & 127;
        float v = M_s[j * MAT_LD + i] * dinv_s[i] * dinv_s[j];
        if (i == j) v = dinv_s[j] * dinv_s[j];
        M_s[j * MAT_LD + i] = v;
    }
    __syncthreads();

    // ---- Stage 5: T = X @ W1   (128x32 @ 32x128) --------------------------
    strip_mm16x4<false>(obs_s, OBS_LD, W1, HID, T_s, MAT_LD, nullptr, D_IN,
                        tm, tn0, lane);
    __syncthreads();

    // ---- Stage 6: H = tanh(M @ T + b1) ------------------------------------
    strip_mm16x4<true>(M_s, MAT_LD, T_s, MAT_LD, H_s, MAT_LD, b1, HID,
                       tm, tn0, lane);
    __syncthreads();

    // ---- Stage 7: T = H @ W2   (128x128 @ 128x128) ------------------------
    strip_mm16x4<false>(H_s, MAT_LD, W2, HID, T_s, MAT_LD, nullptr, HID,
                        tm, tn0, lane);
    __syncthreads();

    // ---- Stage 8: H = tanh(M @ T + b2) ------------------------------------
    strip_mm16x4<true>(M_s, MAT_LD, T_s, MAT_LD, H_s, MAT_LD, b2, HID,
                       tm, tn0, lane);
    __syncthreads();

    // ---- Stage 9: out = H @ Wout + bout (4 lanes/agent, shuffle reduce) ---
    {
        const int a  = tid >> 2;
        const int sl = tid & 3;
        const float* hrow = H_s + a * MAT_LD + sl * 32;
        const float* wseg = Wout + sl * 32;
        float s = 0.0f;
        for (int k = 0; k < 32; ++k) s += hrow[k] * wseg[k];
        s += __shfl_xor(s, 1);
        s += __shfl_xor(s, 2);
        if (sl == 0) out[(size_t)bidx * A_AGENTS + a] = s + bout[0];
    }
}

extern "C" void kernel_launch(void* const* d_in, const int* in_sizes, int n_in,
                              void* d_out, int out_size, void* d_ws, size_t ws_size,
                              hipStream_t stream) {
    (void)n_in; (void)out_size; (void)d_ws; (void)ws_size;
    const float* obs  = (const float*)d_in[0];
    const float* W1   = (const float*)d_in[1];
    const float* b1   = (const float*)d_in[2];
    const float* W2   = (const float*)d_in[3];
    const float* b2   = (const float*)d_in[4];
    const float* Wout = (const float*)d_in[5];
    const float* bout = (const float*)d_in[6];
    float* out = (float*)d_out;

    const int B = in_sizes[0] / (A_AGENTS * D_IN);   // 512

    static_assert(LDS_BYTES <= 320 * 1024, "LDS over WGP capacity");
    (void)hipFuncSetAttribute((const void*)gnn_critic_fused,
                              hipFuncAttributeMaxDynamicSharedMemorySize,
                              LDS_BYTES);
    gnn_critic_fused<<<B, NTHREADS, LDS_BYTES, stream>>>(
        obs, W1, b1, W2, b2, Wout, bout, out);
}